// JitterLayer_46420006535471
// MI455X (gfx1250) — compile-verified
//
#include <hip/hip_runtime.h>
#include <stdint.h>

#ifndef __has_builtin
#define __has_builtin(x) 0
#endif

#define BATCH   64
#define TLEN    2048
#define DDIM    80
#define ROWS    32            // rows of output per block tile
#define LSTR    84            // LDS row stride in floats (bank-conflict + 16B align friendly)
#define THREADS 320           // 10 waves = 80 cols x 4 row strips
#define NTILES  (TLEN / ROWS) // 64
#define NBLOCKS (BATCH * NTILES) // 4096
#define NTOTAL  (64.0 * 2048.0 * 80.0)

// ---- CDNA5 async global->LDS path (guarded; falls back to plain copy) ----
#if defined(__AMDGCN__) && __has_builtin(__builtin_amdgcn_global_load_async_to_lds_b128) && __has_builtin(__builtin_amdgcn_s_wait_asynccnt)
#define USE_ASYNC 1
#else
#define USE_ASYNC 0
#endif

#if USE_ASYNC
// Builtin expects pointers to 16-byte int vectors: arg0 in addrspace(1)
// (prints as "__device__"), arg1 the LDS destination in addrspace(3).
typedef __attribute__((__vector_size__(4 * sizeof(int)))) int v4i_t;
typedef __attribute__((address_space(1))) v4i_t* gptr_v4i;
typedef __attribute__((address_space(3))) v4i_t* lptr_v4i;
#endif

__device__ __forceinline__ float fast_exp2(float x) {
#if defined(__AMDGCN__) && __has_builtin(__builtin_amdgcn_exp2f)
  return __builtin_amdgcn_exp2f(x);      // v_exp_f32
#else
  return exp2f(x);
#endif
}

__device__ __forceinline__ float fast_log2(float x) {
#if defined(__AMDGCN__) && __has_builtin(__builtin_amdgcn_logf)
  return __builtin_amdgcn_logf(x);       // v_log_f32 (log2)
#else
  return log2f(x);
#endif
}

// 3-wide fetch of one tgt row from the LDS tile with row/col zero-padding.
__device__ __forceinline__ void fetch3(const float* t, int tr, int grow, int j,
                                       float& a, float& m, float& z) {
  const bool rv = (grow >= 0) && (grow < TLEN);
  const int base = tr * LSTR + j;
  const float va = t[base - (j > 0 ? 1 : 0)];
  const float vm = t[base];
  const float vz = t[base + (j < DDIM - 1 ? 1 : 0)];
  a = (rv && (j > 0))        ? va : 0.0f;
  m = rv                     ? vm : 0.0f;
  z = (rv && (j < DDIM - 1)) ? vz : 0.0f;
}

__global__ __launch_bounds__(THREADS)
void jitter_main(const float* __restrict__ inp, const float* __restrict__ tgt,
                 float* __restrict__ partial) {
  __shared__ float tile[(ROWS + 2) * LSTR];   // 34 tgt rows (tile + halo)
  __shared__ float redC[THREADS];
  __shared__ float redS[THREADS];

  const int blk = blockIdx.x;       // 0..4095
  const int b   = blk >> 6;         // batch
  const int rt  = blk & 63;         // row-tile within batch
  const int r0  = rt * ROWS;
  const int tid = threadIdx.x;

  const float* __restrict__ tgtBase = tgt + (size_t)b * TLEN * DDIM;
  const float* __restrict__ inpBase = inp + (size_t)b * TLEN * DDIM;

  // Warm the streamed input path (global_prefetch_b8 on gfx1250).
  __builtin_prefetch(inpBase + (size_t)r0 * DDIM + tid, 0, 3);

  // ---- Stage tgt rows [r0-1, r0+32] into LDS (16B-aligned b128 chunks) ----
  for (int chunk = tid; chunk < (ROWS + 2) * (DDIM / 4); chunk += THREADS) {
    const int lrow = chunk / (DDIM / 4);
    const int cv   = chunk % (DDIM / 4);
    const int g    = r0 - 1 + lrow;
    if (g >= 0 && g < TLEN) {
      const float* src = tgtBase + (size_t)g * DDIM + cv * 4;
      float*       dst = &tile[lrow * LSTR + cv * 4];
#if USE_ASYNC
      __builtin_amdgcn_global_load_async_to_lds_b128(
          (gptr_v4i)(size_t)src,
          (lptr_v4i)(uint32_t)(size_t)dst,
          0, 0);
#else
      *(float4*)dst = *(const float4*)src;
#endif
    }
  }
#if USE_ASYNC
  __builtin_amdgcn_s_wait_asynccnt(0);
#endif
  __syncthreads();

  // ---- Compute: col j, 4 strips of 8 rows; rolling 3x3 register window ----
  const int j     = tid % DDIM;
  const int strip = tid / DDIM;   // 0..3
  const int lb    = strip * 8;    // first local row of this strip

  const float K2   = 46.166241308446828f;    // 32 * log2(e)
  const float LN2K = 0.021660849392498293f;  // ln(2) / 32

  float accC = 0.0f, accS = 0.0f;
  float p0, p1, p2, c0, c1, c2, n0, n1, n2;
  fetch3(tile, lb,     r0 + lb - 1, j, p0, p1, p2);
  fetch3(tile, lb + 1, r0 + lb,     j, c0, c1, c2);

  const float* __restrict__ xrow = inpBase + (size_t)(r0 + lb) * DDIM + j;
#pragma unroll
  for (int k = 0; k < 8; ++k) {
    const int gr = r0 + lb + k;
    fetch3(tile, lb + 2 + k, gr + 1, j, n0, n1, n2);
    const float x = xrow[(size_t)k * DDIM];

    const float d0 = fabsf(x - c1);   // center (shift 0)
    const float d1 = fabsf(x - p0);
    const float d2 = fabsf(x - p1);
    const float d3 = fabsf(x - p2);
    const float d4 = fabsf(x - c0);
    const float d5 = fabsf(x - c2);
    const float d6 = fabsf(x - n0);
    const float d7 = fabsf(x - n1);
    const float d8 = fabsf(x - n2);

    const float mn = fminf(d0,
                     fminf(fminf(fminf(d1, d2), fminf(d3, d4)),
                           fminf(fminf(d5, d6), fminf(d7, d8))));

    float s = fast_exp2((mn - d0) * K2);
    s += fast_exp2((mn - d1) * K2);
    s += fast_exp2((mn - d2) * K2);
    s += fast_exp2((mn - d3) * K2);
    s += fast_exp2((mn - d4) * K2);
    s += fast_exp2((mn - d5) * K2);
    s += fast_exp2((mn - d6) * K2);
    s += fast_exp2((mn - d7) * K2);
    s += fast_exp2((mn - d8) * K2);

    const float sm = mn - fast_log2(s) * LN2K;

    accC += d0;
    accS += sm;

    p0 = c0; p1 = c1; p2 = c2;
    c0 = n0; c1 = n1; c2 = n2;
  }

  // ---- Deterministic block tree reduction ----
  redC[tid] = accC;
  redS[tid] = accS;
  __syncthreads();
  if (tid < 64) { redC[tid] += redC[tid + 256]; redS[tid] += redS[tid + 256]; }
  __syncthreads();
  for (int s = 128; s > 0; s >>= 1) {
    if (tid < s) { redC[tid] += redC[tid + s]; redS[tid] += redS[tid + s]; }
    __syncthreads();
  }
  if (tid == 0) {
    partial[2 * blk + 0] = redC[0];
    partial[2 * blk + 1] = redS[0];
  }
}

__global__ __launch_bounds__(256)
void jitter_reduce(const float* __restrict__ partial, float* __restrict__ out) {
  __shared__ double sc[256];
  __shared__ double ss[256];
  double ac = 0.0, as = 0.0;
  for (int i = threadIdx.x; i < NBLOCKS; i += 256) {
    ac += (double)partial[2 * i + 0];
    as += (double)partial[2 * i + 1];
  }
  sc[threadIdx.x] = ac;
  ss[threadIdx.x] = as;
  __syncthreads();
  for (int s = 128; s > 0; s >>= 1) {
    if (threadIdx.x < s) {
      sc[threadIdx.x] += sc[threadIdx.x + s];
      ss[threadIdx.x] += ss[threadIdx.x + s];
    }
    __syncthreads();
  }
  if (threadIdx.x == 0) {
    const double invN = 1.0 / NTOTAL;
    out[0] = (float)(0.5 * (sc[0] * invN + ss[0] * invN));
  }
}

extern "C" void kernel_launch(void* const* d_in, const int* in_sizes, int n_in,
                              void* d_out, int out_size, void* d_ws, size_t ws_size,
                              hipStream_t stream) {
  (void)in_sizes; (void)n_in; (void)out_size; (void)ws_size;
  const float* inp = (const float*)d_in[0];
  const float* tgt = (const float*)d_in[1];
  float* out     = (float*)d_out;
  float* partial = (float*)d_ws;   // 2 * 4096 floats = 32 KB scratch

  jitter_main<<<NBLOCKS, THREADS, 0, stream>>>(inp, tgt, partial);
  jitter_reduce<<<1, 256, 0, stream>>>(partial, out);
}